// GraphSAGEModel_13804024889624
// MI455X (gfx1250) — compile-verified
//
#include <hip/hip_runtime.h>
#include <math.h>

#define D 64

typedef __attribute__((ext_vector_type(2))) float v2f;
typedef __attribute__((ext_vector_type(8))) float v8f;

// ---------------------------------------------------------------------------
// Degree: deg[dst[e]] += 1
// ---------------------------------------------------------------------------
__global__ void deg_kernel(const int* __restrict__ dst, float* __restrict__ deg, int E) {
    int e = blockIdx.x * blockDim.x + threadIdx.x;
    if (e < E) atomicAdd(&deg[dst[e]], 1.0f);
}

// ---------------------------------------------------------------------------
// Scatter: msg[dst[e]][d] += feat[src[e]][d]   (one thread per (e,d))
// ---------------------------------------------------------------------------
__global__ void scatter_kernel(const float* __restrict__ feat, const int* __restrict__ src,
                               const int* __restrict__ dst, float* __restrict__ msg, int E) {
    long long t = (long long)blockIdx.x * blockDim.x + threadIdx.x;
    if (t >= (long long)E * D) return;
    int e = (int)(t >> 6);
    int d = (int)(t & (D - 1));
    atomicAdd(&msg[(long long)dst[e] * D + d], feat[(long long)src[e] * D + d]);
}

// ---------------------------------------------------------------------------
// h_neigh = msg / max(deg, 1)
// ---------------------------------------------------------------------------
__global__ void hdiv_kernel(float* __restrict__ msg, const float* __restrict__ deg, int N) {
    int t = blockIdx.x * blockDim.x + threadIdx.x;
    if (t < N * D) msg[t] = msg[t] / fmaxf(deg[t >> 6], 1.0f);
}

// ---------------------------------------------------------------------------
// out = relu([Aself | Aneigh] @ [Wself ; Wneigh] + b)
// Fused K=128 GEMM via V_WMMA_F32_16X16X4_F32.
// Block = 128 threads = 4 waves; wave w computes rows [blk*64 + w*16, +16) x 64 cols.
// ---------------------------------------------------------------------------
#define BSTRIDE 72  // LDS row stride: 2*72 % 64 == 16 -> half-waves hit disjoint banks

__global__ __launch_bounds__(128) void sage_gemm(
    const float* __restrict__ Aself, const float* __restrict__ Aneigh,
    const float* __restrict__ Wself, const float* __restrict__ Wneigh,
    const float* __restrict__ bias, float* __restrict__ out, int N) {
    __shared__ float Bs[128 * BSTRIDE];  // rows 0..63 = Wself, 64..127 = Wneigh
    __shared__ float bsh[D];

    int tid = threadIdx.x;
    for (int i = tid; i < D * D; i += 128) {
        int r = i >> 6, c = i & (D - 1);
        Bs[r * BSTRIDE + c]         = Wself[i];
        Bs[(r + D) * BSTRIDE + c]   = Wneigh[i];
    }
    if (tid < D) bsh[tid] = bias[tid];
    __syncthreads();

    const int wave = tid >> 5;
    const int lane = tid & 31;
    const int rowBase = blockIdx.x * 64 + wave * 16;
    const int m  = lane & 15;              // A row within tile  / B & C column within tile
    const int kk = (lane >> 4) << 1;       // half-wave K offset: 0 or 2
    const int row = rowBase + m;
    const bool rowOK = (row < N);
    const int rowc = rowOK ? row : 0;      // clamp so loads are always in-bounds

    v8f c[4] = {{}, {}, {}, {}};

    for (int kb = 0; kb < 2 * D; kb += 4) {
        const float* Arow = (kb < D) ? (Aself + (size_t)rowc * D + kb)
                                     : (Aneigh + (size_t)rowc * D + (kb - D));
        v2f a;
        a.x = Arow[kk];
        a.y = Arow[kk + 1];
        if (!rowOK) { a.x = 0.0f; a.y = 0.0f; }

#pragma unroll
        for (int nt = 0; nt < 4; ++nt) {
            v2f b;
            b.x = Bs[(kb + kk) * BSTRIDE + nt * 16 + m];
            b.y = Bs[(kb + kk + 1) * BSTRIDE + nt * 16 + m];
            c[nt] = __builtin_amdgcn_wmma_f32_16x16x4_f32(
                false, a, false, b, (short)0, c[nt], false, false);
        }
    }

    // C layout: VGPR r -> rows r (lanes 0-15) / r+8 (lanes 16-31), col = lane&15
    const int mst = (lane < 16) ? 0 : 8;
#pragma unroll
    for (int r = 0; r < 8; ++r) {
        int orow = rowBase + mst + r;
        if (orow < N) {
#pragma unroll
            for (int nt = 0; nt < 4; ++nt) {
                int col = nt * 16 + m;
                float v = c[nt][r] + bsh[col];
                out[(size_t)orow * D + col] = fmaxf(v, 0.0f);
            }
        }
    }
}

// ---------------------------------------------------------------------------
// out[e] = sigmoid(sigmoid(dot(x[src[e]], x[dst[e]])))   (16 lanes per edge)
// ---------------------------------------------------------------------------
__global__ void edge_dot_kernel(const float* __restrict__ x, const int* __restrict__ src,
                                const int* __restrict__ dst, float* __restrict__ out, int E) {
    long long t = (long long)blockIdx.x * blockDim.x + threadIdx.x;
    int e = (int)(t >> 4);
    int l = (int)(t & 15);
    if (e >= E) return;
    const float4* a = (const float4*)(x + (long long)src[e] * D);
    const float4* b = (const float4*)(x + (long long)dst[e] * D);
    float4 av = a[l], bv = b[l];
    float p = av.x * bv.x + av.y * bv.y + av.z * bv.z + av.w * bv.w;
    p += __shfl_xor(p, 8);
    p += __shfl_xor(p, 4);
    p += __shfl_xor(p, 2);
    p += __shfl_xor(p, 1);
    if (l == 0) {
        float s = 1.0f / (1.0f + __expf(-p));
        out[e] = 1.0f / (1.0f + __expf(-s));
    }
}

// ---------------------------------------------------------------------------
extern "C" void kernel_launch(void* const* d_in, const int* in_sizes, int n_in,
                              void* d_out, int out_size, void* d_ws, size_t ws_size,
                              hipStream_t stream) {
    const float* feat  = (const float*)d_in[0];
    const int*   src   = (const int*)d_in[1];
    const int*   dst   = (const int*)d_in[2];
    const float* Ws1   = (const float*)d_in[3];
    const float* Wn1   = (const float*)d_in[4];
    const float* b1    = (const float*)d_in[5];
    const float* Ws2   = (const float*)d_in[6];
    const float* Wn2   = (const float*)d_in[7];
    const float* b2    = (const float*)d_in[8];
    float* out = (float*)d_out;

    const int N = in_sizes[0] / D;
    const int E = in_sizes[1];

    float* msg = (float*)d_ws;              // N*D
    float* deg = msg + (size_t)N * D;       // N
    float* x1  = deg + N;                   // N*D
    float* x2  = x1 + (size_t)N * D;        // N*D

    const long long ed = (long long)E * D;
    dim3 blk(256);
    dim3 gScat((unsigned)((ed + 255) / 256));
    dim3 gEdge((unsigned)((E + 255) / 256));
    dim3 gNode((unsigned)(((long long)N * D + 255) / 256));
    dim3 gDot((unsigned)(((long long)E * 16 + 255) / 256));
    dim3 gGemm((unsigned)((N + 63) / 64));

    // ---- layer 1 ----
    hipMemsetAsync(msg, 0, (size_t)N * D * sizeof(float), stream);
    hipMemsetAsync(deg, 0, (size_t)N * sizeof(float), stream);
    deg_kernel<<<gEdge, blk, 0, stream>>>(dst, deg, E);
    scatter_kernel<<<gScat, blk, 0, stream>>>(feat, src, dst, msg, E);
    hdiv_kernel<<<gNode, blk, 0, stream>>>(msg, deg, N);
    sage_gemm<<<gGemm, 128, 0, stream>>>(feat, msg, Ws1, Wn1, b1, x1, N);

    // ---- layer 2 (deg unchanged, reuse) ----
    hipMemsetAsync(msg, 0, (size_t)N * D * sizeof(float), stream);
    scatter_kernel<<<gScat, blk, 0, stream>>>(x1, src, dst, msg, E);
    hdiv_kernel<<<gNode, blk, 0, stream>>>(msg, deg, N);
    sage_gemm<<<gGemm, 128, 0, stream>>>(x1, msg, Ws2, Wn2, b2, x2, N);

    // ---- edge dot head ----
    edge_dot_kernel<<<gDot, blk, 0, stream>>>(x2, src, dst, out, E);
}